// Group_42348377538665
// MI455X (gfx1250) — compile-verified
//
#include <hip/hip_runtime.h>

// ---------------------------------------------------------------------------
// Point-cloud grouping (FPS + kNN) for MI455X (gfx1250, wave32).
//   xyz: (16, 8192, 3) f32
//   outputs: neighborhood (16,512,32,3) then center (16,512,3), flat in d_out.
// ---------------------------------------------------------------------------

#define B_      16
#define N_      8192
#define NG_     512
#define GS_     32
#define FPS_T   1024        // 32 wave32 per FPS block
#define FPS_W   (FPS_T/32)  // waves per FPS block
#define GRP_T   256         // 8 waves per grouping block
#define CHUNK_  256         // points processed per chunk in grouping kernel

typedef float v2f __attribute__((ext_vector_type(2)));
typedef float v8f __attribute__((ext_vector_type(8)));

// argmax combine: larger value wins; tie -> smaller point index (jnp.argmax)
__device__ __forceinline__ void amax_combine(float& v, int& i, float ov, int oi) {
  if (ov > v || (ov == v && oi < i)) { v = ov; i = oi; }
}

// ---------------------------------------------------------------------------
// Kernel 1: farthest point sampling. One block per batch; 511 serial steps.
// Per-thread state (8 points + 8 running min-dists) lives in VGPRs. Argmax is
// a 5-step wave32 shuffle butterfly + one 32-entry LDS stage: only 2 block
// barriers per serial step (vs 11 for a pure LDS tree).
// ---------------------------------------------------------------------------
__global__ __launch_bounds__(FPS_T)
void fps_kernel(const float* __restrict__ xyz, float* __restrict__ centers) {
  const int b    = blockIdx.x;
  const int tid  = threadIdx.x;
  const int lane = tid & 31;
  const int wave = tid >> 5;
  const float* P = xyz + (size_t)b * N_ * 3;
  float* C = centers + (size_t)b * NG_ * 3;

  __shared__ float s_wv[FPS_W];
  __shared__ int   s_wi[FPS_W];
  __shared__ float s_last[3];

  // Cache this thread's 8 points in registers (24 VGPRs).
  float px[8], py[8], pz[8], dmin[8];
  const float INF = __builtin_inff();
#pragma unroll
  for (int i = 0; i < 8; ++i) {
    const int n = tid * 8 + i;
    px[i] = P[n * 3 + 0];
    py[i] = P[n * 3 + 1];
    pz[i] = P[n * 3 + 2];
    dmin[i] = INF;
  }
  if (tid == 0) {
    s_last[0] = P[0]; s_last[1] = P[1]; s_last[2] = P[2];
    C[0] = P[0]; C[1] = P[1]; C[2] = P[2];   // center 0 = point 0
  }

  // Reference scan: idx[s] = curr before update; final argmax is discarded.
  // => 511 productive steps, each selecting center s from distances to
  //    center s-1.
  for (int s = 1; s < NG_; ++s) {
    __syncthreads();                       // s_last from previous step visible
    const float lx = s_last[0], ly = s_last[1], lz = s_last[2];

    // update running min-dists; thread-local argmax (first max = lowest idx)
    float bv = -INF; int bi = tid * 8;
#pragma unroll
    for (int i = 0; i < 8; ++i) {
      const float dx = px[i] - lx, dy = py[i] - ly, dz = pz[i] - lz;
      const float d  = dx * dx + dy * dy + dz * dz;
      const float dd = fminf(dmin[i], d);
      dmin[i] = dd;
      if (dd > bv) { bv = dd; bi = tid * 8 + i; }
    }

    // wave32 butterfly argmax (no barriers)
#pragma unroll
    for (int off = 16; off > 0; off >>= 1) {
      const float ov = __shfl_xor(bv, off, 32);
      const int   oi = __shfl_xor(bi, off, 32);
      amax_combine(bv, bi, ov, oi);
    }
    if (lane == 0) { s_wv[wave] = bv; s_wi[wave] = bi; }
    __syncthreads();

    // wave 0 reduces the 32 per-wave winners and publishes the next center
    if (wave == 0) {
      float v = s_wv[lane]; int i = s_wi[lane];
#pragma unroll
      for (int off = 16; off > 0; off >>= 1) {
        const float ov = __shfl_xor(v, off, 32);
        const int   oi = __shfl_xor(i, off, 32);
        amax_combine(v, i, ov, oi);
      }
      if (lane == 0) {
        const float nx = P[i * 3 + 0], ny = P[i * 3 + 1], nz = P[i * 3 + 2];
        s_last[0] = nx; s_last[1] = ny; s_last[2] = nz;
        C[s * 3 + 0] = nx; C[s * 3 + 1] = ny; C[s * 3 + 2] = nz;
      }
    }
  }
}

// ---------------------------------------------------------------------------
// Kernel 2: fused distance (WMMA f32 16x16x4) + top-32 + gather/subtract.
// Block = 8 waves, handles 16 centers of one batch. d2 tile trick:
//   A row m = (cx, cy, cz, 1)     (centers)
//   B col n = (-2px, -2py, -2pz, |p|^2)
//   A*B = |p|^2 - 2 c.p ; then add |c|^2 per row.
// Fragment layouts per CDNA5 ISA (32-bit 16x4 A): lanes 0-15 hold K=0,1 in
// v[0],v[1]; lanes 16-31 hold K=2,3. B mirrored.
// ---------------------------------------------------------------------------
__global__ __launch_bounds__(GRP_T)
void group_kernel(const float* __restrict__ xyz,
                  const float* __restrict__ centers,
                  float* __restrict__ neigh) {
  const int b   = blockIdx.y;
  const int g0  = blockIdx.x * 16;
  const int tid = threadIdx.x;
  const float* P = xyz     + (size_t)b * N_ * 3;
  const float* C = centers + (size_t)b * NG_ * 3;

  __shared__ float s_cx[16], s_cy[16], s_cz[16], s_c2[16];
  __shared__ float s_d2[16][CHUNK_ + 1];   // +1: kill bank conflicts
  __shared__ float s_topd[16][GS_ + 1];
  __shared__ int   s_topi[16][GS_ + 1];

  if (tid < 16) {
    const float cx = C[(g0 + tid) * 3 + 0];
    const float cy = C[(g0 + tid) * 3 + 1];
    const float cz = C[(g0 + tid) * 3 + 2];
    s_cx[tid] = cx; s_cy[tid] = cy; s_cz[tid] = cz;
    s_c2[tid] = cx * cx + cy * cy + cz * cz;
  }
  __syncthreads();

  const int lane = tid & 31;
  const int wave = tid >> 5;
  const int half = lane >> 4;       // 0: lanes 0-15 (K=0,1), 1: lanes 16-31 (K=2,3)
  const int l15  = lane & 15;

  // A fragment (constant across chunks): row m = l15.
  v2f a;
  a.x = half ? s_cz[l15] : s_cx[l15];   // K=2 : K=0
  a.y = half ? 1.0f      : s_cy[l15];   // K=3 : K=1

  // Streaming top-32 state (meaningful only for tid < 16).
  int   filled = 0, maxpos = 0;
  float curmax = -__builtin_inff();

  for (int chunk = 0; chunk < N_ / CHUNK_; ++chunk) {
    const int base = chunk * CHUNK_;
    // each wave computes 2 of the 16 tiles (16 points each) of this chunk
#pragma unroll
    for (int t = 0; t < 2; ++t) {
      const int tile = wave * 2 + t;
      const int pn   = base + tile * 16 + l15;
      const float qx = P[pn * 3 + 0];
      const float qy = P[pn * 3 + 1];
      const float qz = P[pn * 3 + 2];
      v2f bb;
      bb.x = half ? (-2.0f * qz) : (-2.0f * qx);
      bb.y = half ? (qx * qx + qy * qy + qz * qz) : (-2.0f * qy);
      v8f acc = {};
      acc = __builtin_amdgcn_wmma_f32_16x16x4_f32(
          /*neg_a=*/false, a, /*neg_b=*/false, bb,
          /*c_mod=*/(short)0, acc, /*reuse_a=*/false, /*reuse_b=*/false);
      // C/D layout: VGPR v, lanes 0-15 -> M=v, lanes 16-31 -> M=v+8; N=l15
#pragma unroll
      for (int v = 0; v < 8; ++v) {
        const int m = v + 8 * half;
        s_d2[m][tile * 16 + l15] = acc[v] + s_c2[m];
      }
    }
    __syncthreads();

    if (tid < 16) {                  // one selection thread per center
      for (int j = 0; j < CHUNK_; ++j) {
        const float d    = s_d2[tid][j];
        const int   pidx = base + j;
        if (filled < GS_) {
          s_topd[tid][filled] = d; s_topi[tid][filled] = pidx;
          if (++filled == GS_) {
            curmax = -__builtin_inff();
            for (int k = 0; k < GS_; ++k)
              if (s_topd[tid][k] >= curmax) { curmax = s_topd[tid][k]; maxpos = k; }
          }
        } else if (d < curmax) {     // strict <: equal dists keep earlier index
          s_topd[tid][maxpos] = d; s_topi[tid][maxpos] = pidx;
          curmax = -__builtin_inff();
          for (int k = 0; k < GS_; ++k)   // >= picks latest (largest-index) max
            if (s_topd[tid][k] >= curmax) { curmax = s_topd[tid][k]; maxpos = k; }
        }
      }
    }
    __syncthreads();
  }

  // Sort ascending by (d2, idx) to match jax.lax.top_k order.
  if (tid < 16) {
    for (int i = 1; i < GS_; ++i) {
      const float d  = s_topd[tid][i];
      const int   ix = s_topi[tid][i];
      int j = i - 1;
      while (j >= 0 && (s_topd[tid][j] > d ||
                        (s_topd[tid][j] == d && s_topi[tid][j] > ix))) {
        s_topd[tid][j + 1] = s_topd[tid][j];
        s_topi[tid][j + 1] = s_topi[tid][j];
        --j;
      }
      s_topd[tid][j + 1] = d; s_topi[tid][j + 1] = ix;
    }
  }
  __syncthreads();

  // neighborhood = gathered points - center
  for (int e = tid; e < 16 * GS_; e += GRP_T) {
    const int m = e >> 5, k = e & 31;
    const int pidx = s_topi[m][k];
    const size_t o = (((size_t)b * NG_ + (g0 + m)) * GS_ + k) * 3;
    neigh[o + 0] = P[pidx * 3 + 0] - s_cx[m];
    neigh[o + 1] = P[pidx * 3 + 1] - s_cy[m];
    neigh[o + 2] = P[pidx * 3 + 2] - s_cz[m];
  }
}

// ---------------------------------------------------------------------------
extern "C" void kernel_launch(void* const* d_in, const int* in_sizes, int n_in,
                              void* d_out, int out_size, void* d_ws, size_t ws_size,
                              hipStream_t stream) {
  (void)in_sizes; (void)n_in; (void)out_size; (void)d_ws; (void)ws_size;
  const float* xyz = (const float*)d_in[0];
  float* out     = (float*)d_out;
  float* neigh   = out;                                  // 16*512*32*3
  float* centers = out + (size_t)B_ * NG_ * GS_ * 3;     // 16*512*3

  fps_kernel<<<B_, FPS_T, 0, stream>>>(xyz, centers);
  group_kernel<<<dim3(NG_ / 16, B_), GRP_T, 0, stream>>>(xyz, centers, neigh);
}